// GNNModel_20916490731553
// MI455X (gfx1250) — compile-verified
//
#include <hip/hip_runtime.h>
#include <hip/hip_bf16.h>

#define NNODES 50000
#define NEDGES 1600000
#define FDIM   128
#define NGRAPH 512
#define NCLASS 2

typedef __attribute__((ext_vector_type(2))) float v2f;
typedef __attribute__((ext_vector_type(8))) float v8f;

// ---------------------------------------------------------------------------
// Degree prep
// ---------------------------------------------------------------------------
__global__ void k_init_deg(float* deg) {
    int i = blockIdx.x * 256 + threadIdx.x;
    if (i < NNODES) deg[i] = 1.0f;   // self-loop term
}

__global__ void k_accum_deg(const int* __restrict__ dst, float* deg) {
    int e = blockIdx.x * 256 + threadIdx.x;
    if (e < NEDGES) atomicAdd(&deg[dst[e]], 1.0f);
}

__global__ void k_rsqrt(float* deg_to_disqrt) {
    int i = blockIdx.x * 256 + threadIdx.x;
    if (i < NNODES) deg_to_disqrt[i] = rsqrtf(deg_to_disqrt[i]);
}

// ---------------------------------------------------------------------------
// WMMA fp32 GEMM:  Out[rows x 128] = X[rows x 128] @ W[128 x 128] (+bias, relu)
// One block = 16 rows; 8 waves each own a 16-wide column tile.
// rows must be a multiple of 16 (50000 and 512 both are).
// ---------------------------------------------------------------------------
#define LDS_STRIDE 129   // pad to kill 16-way bank conflicts on A reads

__global__ __launch_bounds__(256) void k_gemm_wmma_f32(
    const float* __restrict__ X, const float* __restrict__ Wm,
    const float* __restrict__ bias, float* __restrict__ Out, int do_relu)
{
    __shared__ float tileA[16 * LDS_STRIDE];
    const int tid  = threadIdx.x;
    const int wave = tid >> 5;          // 0..7  -> column tile
    const int lane = tid & 31;
    const long rowBase = (long)blockIdx.x * 16;

    // Cooperative stage of the 16x128 A tile into LDS (padded stride).
    const float* src = X + rowBase * FDIM;
    for (int i = tid; i < 16 * FDIM; i += 256) {
        int r = i >> 7, c = i & 127;
        tileA[r * LDS_STRIDE + c] = src[i];
    }
    __syncthreads();

    const int colBase = wave * 16;
    const int half = lane >> 4;         // 0: K pair {k,k+1}, 1: {k+2,k+3}
    const int l15  = lane & 15;

    v8f acc = {};
    #pragma unroll 4
    for (int k = 0; k < FDIM; k += 4) {
        const int ka = k + half * 2;
        v2f a, b;
        // A fragment: 16x4, lanes 0-15 hold K={k,k+1}, lanes 16-31 K={k+2,k+3}
        a.x = tileA[l15 * LDS_STRIDE + ka + 0];
        a.y = tileA[l15 * LDS_STRIDE + ka + 1];
        // B fragment: 4x16, same K split, lane%16 selects column
        b.x = Wm[(ka + 0) * FDIM + colBase + l15];
        b.y = Wm[(ka + 1) * FDIM + colBase + l15];
        acc = __builtin_amdgcn_wmma_f32_16x16x4_f32(
                  false, a, false, b, (short)0, acc, false, false);
    }

    // C/D layout: VGPR v -> row (v + half*8), N = lane%16
    const float bcol = bias ? bias[colBase + l15] : 0.0f;
    #pragma unroll
    for (int v = 0; v < 8; ++v) {
        float val = acc[v] + bcol;
        if (do_relu) val = fmaxf(val, 0.0f);
        Out[(rowBase + v + half * 8) * FDIM + colBase + l15] = val;
    }
}

// ---------------------------------------------------------------------------
// agg[n][f] = h[n][f] * self_norm[n] + bias[f]      (self_norm = disqrt^2)
// ---------------------------------------------------------------------------
__global__ void k_init_agg(const float* __restrict__ h,
                           const float* __restrict__ disqrt,
                           const float* __restrict__ bias,
                           float* __restrict__ agg)
{
    long i = (long)blockIdx.x * 256 + threadIdx.x;   // over N*128
    long n = i >> 7;
    int  f = (int)(i & 127);
    float d = disqrt[n];
    agg[i] = h[i] * (d * d) + bias[f];
}

// ---------------------------------------------------------------------------
// Edge scatter: one wave per edge; each lane owns 4 features (float4 gather,
// 4x global_atomic_add_f32 scatter). edge_norm computed on the fly.
// ---------------------------------------------------------------------------
__global__ __launch_bounds__(256) void k_scatter(
    const int* __restrict__ srcI, const int* __restrict__ dstI,
    const float* __restrict__ disqrt,
    const float* __restrict__ h, float* __restrict__ agg)
{
    int e = blockIdx.x * 8 + (threadIdx.x >> 5);
    int lane = threadIdx.x & 31;
    if (e >= NEDGES) return;
    int s = srcI[e], d = dstI[e];
    float w = disqrt[s] * disqrt[d];
    const float4 v = ((const float4*)(h + (long)s * FDIM))[lane];
    float* ad = agg + (long)d * FDIM + lane * 4;
    atomicAdd(ad + 0, v.x * w);
    atomicAdd(ad + 1, v.y * w);
    atomicAdd(ad + 2, v.z * w);
    atomicAdd(ad + 3, v.w * w);
}

// ---------------------------------------------------------------------------
// Per-node L2 normalize + ReLU. One wave per node, shuffle reduction.
// ---------------------------------------------------------------------------
__global__ __launch_bounds__(256) void k_norm_relu(
    const float* __restrict__ agg, float* __restrict__ out)
{
    int n = blockIdx.x * 8 + (threadIdx.x >> 5);
    int lane = threadIdx.x & 31;
    if (n >= NNODES) return;
    const float4 v = ((const float4*)(agg + (long)n * FDIM))[lane];
    float ss = v.x * v.x + v.y * v.y + v.z * v.z + v.w * v.w;
    #pragma unroll
    for (int off = 16; off > 0; off >>= 1)
        ss += __shfl_xor(ss, off, 32);
    float inv = 1.0f / fmaxf(sqrtf(ss), 1e-12f);
    float4 o;
    o.x = fmaxf(v.x * inv, 0.0f);
    o.y = fmaxf(v.y * inv, 0.0f);
    o.z = fmaxf(v.z * inv, 0.0f);
    o.w = fmaxf(v.w * inv, 0.0f);
    ((float4*)(out + (long)n * FDIM))[lane] = o;
}

// ---------------------------------------------------------------------------
// Graph pooling
// ---------------------------------------------------------------------------
__global__ void k_pool_zero(float* pooled) {
    int i = blockIdx.x * 256 + threadIdx.x;
    if (i < NGRAPH * FDIM) pooled[i] = 0.0f;
}

__global__ void k_pool_accum(const float* __restrict__ x,
                             const int* __restrict__ batch,
                             float* __restrict__ pooled)
{
    long i = (long)blockIdx.x * 256 + threadIdx.x;   // over N*128
    long n = i >> 7;
    int  f = (int)(i & 127);
    atomicAdd(&pooled[(long)batch[n] * FDIM + f], x[i]);
}

// ---------------------------------------------------------------------------
// Head: logits[g][c] = dot(z[g,:], lin2_w[:,c]) + lin2_b[c]; one wave/output.
// ---------------------------------------------------------------------------
__global__ __launch_bounds__(256) void k_head(
    const float* __restrict__ z, const float* __restrict__ w2,
    const float* __restrict__ b2, float* __restrict__ out)
{
    int o = blockIdx.x * 8 + (threadIdx.x >> 5);
    int lane = threadIdx.x & 31;
    if (o >= NGRAPH * NCLASS) return;
    int g = o >> 1, c = o & 1;
    const float4 v = ((const float4*)(z + (long)g * FDIM))[lane];
    int k = lane * 4;
    float s = v.x * w2[(k + 0) * NCLASS + c] + v.y * w2[(k + 1) * NCLASS + c]
            + v.z * w2[(k + 2) * NCLASS + c] + v.w * w2[(k + 3) * NCLASS + c];
    #pragma unroll
    for (int off = 16; off > 0; off >>= 1)
        s += __shfl_xor(s, off, 32);
    if (lane == 0) out[o] = s + b2[c];
}

// ---------------------------------------------------------------------------
extern "C" void kernel_launch(void* const* d_in, const int* in_sizes, int n_in,
                              void* d_out, int out_size, void* d_ws, size_t ws_size,
                              hipStream_t stream) {
    const float* x        = (const float*)d_in[0];
    const int*   eidx     = (const int*)d_in[1];         // [2, E]
    const int*   batch    = (const int*)d_in[2];         // [N]
    const float* W0       = (const float*)d_in[3];
    const float* b0       = (const float*)d_in[4];
    const float* W1       = (const float*)d_in[5];
    const float* b1       = (const float*)d_in[6];
    const float* W2       = (const float*)d_in[7];
    const float* b2       = (const float*)d_in[8];
    const float* lin1_w   = (const float*)d_in[9];
    const float* lin1_b   = (const float*)d_in[10];
    const float* lin2_w   = (const float*)d_in[11];
    const float* lin2_b   = (const float*)d_in[12];
    float* logits = (float*)d_out;                        // [G*C]

    const int* esrc = eidx;
    const int* edst = eidx + NEDGES;

    // Workspace layout (floats)
    float* ws    = (float*)d_ws;
    const size_t NF = (size_t)NNODES * FDIM;
    float* bufH   = ws;                    // GEMM output h          [N*128]
    float* bufAgg = bufH + NF;             // aggregation buffer     [N*128]
    float* bufX   = bufAgg + NF;           // layer output / next x  [N*128]
    float* disqrt = bufX + NF;             // 1/sqrt(deg)            [N]
    float* pooled = disqrt + NNODES;       // [G*128]
    float* zbuf   = pooled + (size_t)NGRAPH * FDIM;  // [G*128]

    const int B = 256;
    // 1) degree -> disqrt
    k_init_deg <<<(NNODES + B - 1) / B, B, 0, stream>>>(disqrt);
    k_accum_deg<<<(NEDGES + B - 1) / B, B, 0, stream>>>(edst, disqrt);
    k_rsqrt    <<<(NNODES + B - 1) / B, B, 0, stream>>>(disqrt);

    // 2) three GCN layers
    const float* Ws[3]   = {W0, W1, W2};
    const float* bs[3]   = {b0, b1, b2};
    const float* cur = x;
    const int gemmBlocks   = NNODES / 16;          // 3125 exact
    const int elemBlocks   = (int)(NF / B);        // 25000 exact
    const int edgeBlocks   = NEDGES / 8;           // 200000 exact
    const int nodeBlocks   = NNODES / 8;           // 6250 exact
    for (int l = 0; l < 3; ++l) {
        k_gemm_wmma_f32<<<gemmBlocks, B, 0, stream>>>(cur, Ws[l], nullptr, bufH, 0);
        k_init_agg     <<<elemBlocks, B, 0, stream>>>(bufH, disqrt, bs[l], bufAgg);
        k_scatter      <<<edgeBlocks, B, 0, stream>>>(esrc, edst, disqrt, bufH, bufAgg);
        k_norm_relu    <<<nodeBlocks, B, 0, stream>>>(bufAgg, bufX);
        cur = bufX;
    }

    // 3) pool + MLP head
    k_pool_zero <<<(NGRAPH * FDIM) / B, B, 0, stream>>>(pooled);
    k_pool_accum<<<elemBlocks, B, 0, stream>>>(bufX, batch, pooled);
    k_gemm_wmma_f32<<<NGRAPH / 16, B, 0, stream>>>(pooled, lin1_w, lin1_b, zbuf, 1);
    k_head<<<(NGRAPH * NCLASS) / 8 / (B / 256 * 32) * 32 / 32, B, 0, stream>>>(zbuf, lin2_w, lin2_b, logits);
}